// MVSNet_11209864643161
// MI455X (gfx1250) — compile-verified
//
#include <hip/hip_runtime.h>
#include <math.h>

typedef float v2f __attribute__((ext_vector_type(2)));
typedef float v8f __attribute__((ext_vector_type(8)));

// ---------------------------------------------------------------------------
// Homographies: H[v,d] = K[v] (R_rel[v] + t_rel[v] n^T / depth_d) inv(K[0])
// ---------------------------------------------------------------------------
__global__ void mvs_homog_kernel(const float* __restrict__ K,
                                 const float* __restrict__ E,
                                 float* __restrict__ Hs)
{
    int idx = blockIdx.x * blockDim.x + threadIdx.x;
    if (idx >= 3 * 48) return;
    int v = idx / 48;
    int d = idx - v * 48;
    float depth = 5.0f + 0.1f * (float)d;

    float R[9], R0[9], t[3], t0[3], Kv[9], K0[9];
    for (int r = 0; r < 3; ++r) {
        for (int c = 0; c < 3; ++c) {
            R[r*3+c]  = E[v*16 + r*4 + c];
            R0[r*3+c] = E[r*4 + c];
            Kv[r*3+c] = K[v*9 + r*3 + c];
            K0[r*3+c] = K[r*3 + c];
        }
        t[r]  = E[v*16 + r*4 + 3];
        t0[r] = E[r*4 + 3];
    }
    // R_rel = R * R0^T
    float Rr[9];
    for (int r = 0; r < 3; ++r)
        for (int c = 0; c < 3; ++c)
            Rr[r*3+c] = R[r*3+0]*R0[c*3+0] + R[r*3+1]*R0[c*3+1] + R[r*3+2]*R0[c*3+2];
    // t_rel = t - R_rel * t0
    float tr[3];
    for (int r = 0; r < 3; ++r)
        tr[r] = t[r] - (Rr[r*3+0]*t0[0] + Rr[r*3+1]*t0[1] + Rr[r*3+2]*t0[2]);
    // inv(K0), general adjugate
    float a = K0[0], b = K0[1], c2 = K0[2];
    float dd = K0[3], e = K0[4], f = K0[5];
    float g = K0[6], h = K0[7], i9 = K0[8];
    float det = a*(e*i9 - f*h) - b*(dd*i9 - f*g) + c2*(dd*h - e*g);
    float inv = 1.0f / det;
    float Ki[9];
    Ki[0] =  (e*i9 - f*h) * inv; Ki[1] = -(b*i9 - c2*h) * inv; Ki[2] =  (b*f - c2*e) * inv;
    Ki[3] = -(dd*i9 - f*g) * inv; Ki[4] =  (a*i9 - c2*g) * inv; Ki[5] = -(a*f - c2*dd) * inv;
    Ki[6] =  (dd*h - e*g) * inv; Ki[7] = -(a*h - b*g) * inv;  Ki[8] =  (a*e - b*dd) * inv;
    // M = R_rel with column 2 += t_rel/depth   (n = [0,0,1])
    float M[9];
    for (int x = 0; x < 9; ++x) M[x] = Rr[x];
    float idep = 1.0f / depth;
    M[2] += tr[0] * idep; M[5] += tr[1] * idep; M[8] += tr[2] * idep;
    // H = Kv * M * Ki
    float T[9];
    for (int r = 0; r < 3; ++r)
        for (int c = 0; c < 3; ++c)
            T[r*3+c] = Kv[r*3+0]*M[0*3+c] + Kv[r*3+1]*M[1*3+c] + Kv[r*3+2]*M[2*3+c];
    float* Ho = Hs + (size_t)(v*48 + d) * 9;
    for (int r = 0; r < 3; ++r)
        for (int c = 0; c < 3; ++c)
            Ho[r*3+c] = T[r*3+0]*Ki[0*3+c] + T[r*3+1]*Ki[1*3+c] + T[r*3+2]*Ki[2*3+c];
}

__device__ __forceinline__ int clampi(int x, int lo, int hi) {
    return x < lo ? lo : (x > hi ? hi : x);
}

// ---------------------------------------------------------------------------
// Implicit-GEMM conv via V_WMMA_F32_16X16X4_F32 (full f32 precision).
// One wave computes a 16(out-ch) x 32(pixel) tile = two 16x16 WMMA tiles that
// share each A (weight) fragment. K = ci*kh*kw consumed in chunks of 4.
// Fragment layouts per CDNA5 ISA 7.12.2:
//   A (16x4):  lane L holds row M=L&15, K = 2*(L>>4)+e
//   B (4x16):  lane L holds col N=L&15, K = 2*(L>>4)+e
//   C/D:       lane L elem r = D[r + 8*(L>>4), L&15]
// Halo handling is branch-free: addresses are clamped into the plane (always
// valid), the loaded value is zeroed by select when out of bounds. Invalid
// rows/columns of the tile compute garbage that is never stored (WMMA rows and
// columns are independent).
// MODE 0: BN(g,b,m,v)+ReLU   MODE 1: +bias(p0)
// MODE 2: GRU step with h0=0: out = sigmoid(convZ + p0) * tanh(convQ + p1)
// ---------------------------------------------------------------------------
template<int KSZ, int MODE>
__global__ __launch_bounds__(32)
void mvs_wmma_conv_kernel(const float* __restrict__ in, float* __restrict__ out,
                          const float* __restrict__ w0, const float* __restrict__ w1,
                          const float* __restrict__ p0, const float* __restrict__ p1,
                          const float* __restrict__ p2, const float* __restrict__ p3,
                          int N, int Ci, int Hi, int Wi,
                          int Co, int Ho, int Wo,
                          int stride, int pad, int Keff, int wstride)
{
    constexpr bool GRU = (MODE == 2);
    const int lane = threadIdx.x;
    const int col  = lane & 15;
    const int half = lane >> 4;            // 0: K={0,1}, 1: K={2,3}
    const int klo  = half * 2;
    const int npix = Ho * Wo;
    const int pixBase = blockIdx.x * 32;   // two 16-pixel tiles
    const int coBase  = blockIdx.y * 16;
    const int n = blockIdx.z;
    const int K2 = KSZ * KSZ;

    // A fragment row (clamped; invalid rows produce unstored garbage)
    const int  mrow = coBase + col;
    const int  wrow = (mrow < Co) ? mrow : (Co - 1);
    const float* wr0 = w0 + (size_t)wrow * wstride;
    const float* wr1 = GRU ? (w1 + (size_t)wrow * wstride) : w0;

    // two B-fragment pixel columns (clamped; invalid cols unstored)
    int  opix[2], iyb[2], ixb[2];
    bool pok[2];
#pragma unroll
    for (int t = 0; t < 2; ++t) {
        int op   = pixBase + t * 16 + col;
        pok[t]   = (op < npix);
        int opc  = pok[t] ? op : (npix - 1);
        opix[t]  = opc;
        int oy   = opc / Wo;
        int ox   = opc - oy * Wo;
        iyb[t]   = oy * stride - pad;
        ixb[t]   = ox * stride - pad;
    }

    const float* base = in + (size_t)n * Ci * Hi * Wi;

    v8f accZ[2] = {};
    v8f accQ[2] = {};

    const int Kmain = Keff & ~3;
#pragma unroll 2
    for (int k0 = 0; k0 < Kmain; k0 += 4) {
        v2f a0, a1, bfr[2];
#pragma unroll
        for (int e = 0; e < 2; ++e) {
            int k  = k0 + klo + e;
            a0[e]  = wr0[k];
            a1[e]  = GRU ? wr1[k] : 0.0f;
            int ci  = k / K2;
            int rem = k - ci * K2;
            int ky  = rem / KSZ;
            int kx  = rem - ky * KSZ;
            const float* plane = base + (size_t)ci * Hi * Wi;
#pragma unroll
            for (int t = 0; t < 2; ++t) {
                int iy = iyb[t] + ky;
                int ix = ixb[t] + kx;
                bool ok = (iy >= 0) & (iy < Hi) & (ix >= 0) & (ix < Wi);
                int iyc = clampi(iy, 0, Hi - 1);
                int ixc = clampi(ix, 0, Wi - 1);
                float bv = plane[iyc * Wi + ixc];   // always-valid address
                bfr[t][e] = ok ? bv : 0.0f;         // branch-free mask
            }
        }
        accZ[0] = __builtin_amdgcn_wmma_f32_16x16x4_f32(false, a0, false, bfr[0],
                                                        (short)0, accZ[0], false, false);
        accZ[1] = __builtin_amdgcn_wmma_f32_16x16x4_f32(false, a0, false, bfr[1],
                                                        (short)0, accZ[1], false, false);
        if constexpr (GRU) {
            accQ[0] = __builtin_amdgcn_wmma_f32_16x16x4_f32(false, a1, false, bfr[0],
                                                            (short)0, accQ[0], false, false);
            accQ[1] = __builtin_amdgcn_wmma_f32_16x16x4_f32(false, a1, false, bfr[1],
                                                            (short)0, accQ[1], false, false);
        }
    }

    if (Kmain < Keff) {            // guarded tail chunk (Keff % 4 != 0)
        v2f a0, a1, bfr[2];
#pragma unroll
        for (int e = 0; e < 2; ++e) {
            int k  = Kmain + klo + e;
            int kc = (k < Keff) ? k : (Keff - 1);       // clamped, always valid
            bool kok = (k < Keff);
            float av0 = wr0[kc];
            float av1 = GRU ? wr1[kc] : 0.0f;
            a0[e] = kok ? av0 : 0.0f;
            a1[e] = kok ? av1 : 0.0f;
            int ci  = kc / K2;
            int rem = kc - ci * K2;
            int ky  = rem / KSZ;
            int kx  = rem - ky * KSZ;
            const float* plane = base + (size_t)ci * Hi * Wi;
#pragma unroll
            for (int t = 0; t < 2; ++t) {
                int iy = iyb[t] + ky;
                int ix = ixb[t] + kx;
                bool ok = kok & (iy >= 0) & (iy < Hi) & (ix >= 0) & (ix < Wi);
                int iyc = clampi(iy, 0, Hi - 1);
                int ixc = clampi(ix, 0, Wi - 1);
                float bv = plane[iyc * Wi + ixc];
                bfr[t][e] = ok ? bv : 0.0f;
            }
        }
        accZ[0] = __builtin_amdgcn_wmma_f32_16x16x4_f32(false, a0, false, bfr[0],
                                                        (short)0, accZ[0], false, false);
        accZ[1] = __builtin_amdgcn_wmma_f32_16x16x4_f32(false, a0, false, bfr[1],
                                                        (short)0, accZ[1], false, false);
        if constexpr (GRU) {
            accQ[0] = __builtin_amdgcn_wmma_f32_16x16x4_f32(false, a1, false, bfr[0],
                                                            (short)0, accQ[0], false, false);
            accQ[1] = __builtin_amdgcn_wmma_f32_16x16x4_f32(false, a1, false, bfr[1],
                                                            (short)0, accQ[1], false, false);
        }
    }

    // epilogue: lane holds D[r + 8*half, col] for both pixel tiles
#pragma unroll
    for (int t = 0; t < 2; ++t) {
        if (!pok[t]) continue;
#pragma unroll
        for (int r = 0; r < 8; ++r) {
            int co = coBase + r + half * 8;
            if (co >= Co) continue;
            float val = accZ[t][r];
            if constexpr (MODE == 0) {
                val = (val - p2[co]) * rsqrtf(p3[co] + 1e-5f) * p0[co] + p1[co];
                val = fmaxf(val, 0.0f);
            } else if constexpr (MODE == 1) {
                val += p0[co];
            } else {
                float z = 1.0f / (1.0f + __expf(-(val + p0[co])));
                float q = tanhf(accQ[t][r] + p1[co]);
                val = z * q;
            }
            out[(size_t)(n * Co + co) * npix + opix[t]] = val;
        }
    }
}

// ---------------------------------------------------------------------------
// Warp views 1,2 per depth + variance cost across the 3 views.
// Stores NEGATED cost (reg_net input is -cost).
// ---------------------------------------------------------------------------
__global__ void mvs_warpcost_kernel(const float* __restrict__ feats,
                                    const float* __restrict__ Hs,
                                    float* __restrict__ negcost)
{
    const int Hh = 96, Ww = 128, C = 32, NP = 96 * 128;
    int d   = blockIdx.y;
    int pix = blockIdx.x * blockDim.x + threadIdx.x;
    if (pix >= NP) return;
    int y = pix / Ww, x = pix - y * Ww;
    float fx = (float)x, fy = (float)y;

    int   X0[2], X1[2], Y0[2], Y1[2];
    float W00[2], W01[2], W10[2], W11[2];
    bool  VAL[2];
#pragma unroll
    for (int v = 0; v < 2; ++v) {
        const float* Hm = Hs + (size_t)((v + 1) * 48 + d) * 9;
        float denom = Hm[6]*fx + Hm[7]*fy + Hm[8];
        float rden  = 1.0f / denom;
        float u  = (Hm[0]*fx + Hm[1]*fy + Hm[2]) * rden;
        float vv = (Hm[3]*fx + Hm[4]*fy + Hm[5]) * rden;
        float x0 = floorf(u), y0 = floorf(vv);
        float wx1 = u - x0, wy1 = vv - y0;
        VAL[v] = (u >= 0.0f) && (u <= (float)(Ww - 1)) &&
                 (vv >= 0.0f) && (vv <= (float)(Hh - 1));
        X0[v] = (int)fminf(fmaxf(x0,        0.0f), (float)(Ww - 1));
        X1[v] = (int)fminf(fmaxf(x0 + 1.0f, 0.0f), (float)(Ww - 1));
        Y0[v] = (int)fminf(fmaxf(y0,        0.0f), (float)(Hh - 1));
        Y1[v] = (int)fminf(fmaxf(y0 + 1.0f, 0.0f), (float)(Hh - 1));
        W00[v] = (1.0f - wy1) * (1.0f - wx1);
        W01[v] = (1.0f - wy1) * wx1;
        W10[v] = wy1 * (1.0f - wx1);
        W11[v] = wy1 * wx1;
    }
    const float third = 1.0f / 3.0f;
    for (int c = 0; c < C; ++c) {
        float fr = feats[(size_t)c * NP + pix];   // view 0 (ref)
        float s  = fr, s2 = fr * fr;
#pragma unroll
        for (int v = 0; v < 2; ++v) {
            const float* F = feats + ((size_t)((v + 1) * C + c)) * NP;
            float fv = W00[v] * F[Y0[v]*Ww + X0[v]] + W01[v] * F[Y0[v]*Ww + X1[v]]
                     + W10[v] * F[Y1[v]*Ww + X0[v]] + W11[v] * F[Y1[v]*Ww + X1[v]];
            fv = VAL[v] ? fv : 0.0f;
            s += fv; s2 += fv * fv;
        }
        float mean = s * third;
        negcost[((size_t)(d * C + c)) * NP + pix] = mean * mean - s2 * third;
    }
}

// ---------------------------------------------------------------------------
// Softmax over the 48 depth hypotheses per pixel.
// ---------------------------------------------------------------------------
__global__ void mvs_softmax_kernel(const float* __restrict__ reg,
                                   float* __restrict__ out)
{
    const int NP = 96 * 128, D = 48;
    int pix = blockIdx.x * blockDim.x + threadIdx.x;
    if (pix >= NP) return;
    float vals[D];
    float mx = -3.402823466e+38f;
#pragma unroll
    for (int d = 0; d < D; ++d) {
        float v = reg[(size_t)d * NP + pix];
        vals[d] = v;
        mx = fmaxf(mx, v);
    }
    float s = 0.0f;
#pragma unroll
    for (int d = 0; d < D; ++d) {
        float e = __expf(vals[d] - mx);
        vals[d] = e;
        s += e;
    }
    float rs = 1.0f / s;
#pragma unroll
    for (int d = 0; d < D; ++d)
        out[(size_t)d * NP + pix] = vals[d] * rs;
}

// ---------------------------------------------------------------------------
// Host launcher
// ---------------------------------------------------------------------------
extern "C" void kernel_launch(void* const* d_in, const int* in_sizes, int n_in,
                              void* d_out, int out_size, void* d_ws, size_t ws_size,
                              hipStream_t stream)
{
    (void)in_sizes; (void)n_in; (void)out_size; (void)ws_size;

    // --- unpack inputs (setup_inputs insertion order, recursively flattened) ---
    const float* imgs = (const float*)d_in[0];   // (1,3,3,384,512)
    const float* Kin  = (const float*)d_in[1];   // (3,3,3)
    const float* Ein  = (const float*)d_in[2];   // (3,4,4)
    int p = 3;
    const float *cw[7], *cg[7], *cb[7], *cm[7], *cvv[7];
    for (int l = 0; l < 7; ++l) {
        cw[l]  = (const float*)d_in[p++];
        cg[l]  = (const float*)d_in[p++];
        cb[l]  = (const float*)d_in[p++];
        cm[l]  = (const float*)d_in[p++];
        cvv[l] = (const float*)d_in[p++];
    }
    const float* fw = (const float*)d_in[p++];
    const float* fb = (const float*)d_in[p++];
    const float *gwz[3], *gbz[3], *gwr[3], *gbr[3], *gwq[3], *gbq[3];
    for (int l = 0; l < 3; ++l) {
        gwz[l] = (const float*)d_in[p++];
        gbz[l] = (const float*)d_in[p++];
        gwr[l] = (const float*)d_in[p++];   // dead: h0 = 0 => r*h = 0
        gbr[l] = (const float*)d_in[p++];
        gwq[l] = (const float*)d_in[p++];
        gbq[l] = (const float*)d_in[p++];
    }
    const float* pw = (const float*)d_in[p++];
    const float* pb = (const float*)d_in[p++];
    (void)gwr; (void)gbr;

    // --- workspace layout (float offsets); total 33,621,264 floats ~ 134.5 MB ---
    float* ws = (float*)d_ws;
    float* bufA   = ws + 0;          //  4,718,592  feature ping
    float* bufB   = ws + 4718592;    //  4,718,592  feature pong
    float* g1     = ws + 0;          //  9,437,184  GRU1 out (reuses ping+pong)
    float* feats  = ws + 9437184;    //  1,179,648  (3,32,96,128)
    float* cost   = ws + 10616832;   // 18,874,368  (48,32,96,128) negated cost
    float* g2     = ws + 29491200;   //  2,359,296  (48,4,96,128)
    float* g3     = ws + 31850496;   //  1,179,648  (48,2,96,128)
    float* regout = ws + 33030144;   //    589,824  (48,96,128)
    float* Hs     = ws + 33619968;   //      1,296  (3,48,3,3)

    auto conv = [&](const float* in, float* out, const float* w0, const float* w1,
                    const float* q0, const float* q1, const float* q2, const float* q3,
                    int N, int Ci, int Hi, int Wi, int Co, int Ho, int Wo,
                    int ksz, int stride, int pad, int Keff, int wstride, int mode) {
        dim3 grid((unsigned)((Ho * Wo + 31) / 32), (unsigned)((Co + 15) / 16), (unsigned)N);
        dim3 blk(32);
#define CONV_ARGS in, out, w0, w1, q0, q1, q2, q3, N, Ci, Hi, Wi, Co, Ho, Wo, stride, pad, Keff, wstride
        if (ksz == 5)
            mvs_wmma_conv_kernel<5, 0><<<grid, blk, 0, stream>>>(CONV_ARGS);
        else if (mode == 0)
            mvs_wmma_conv_kernel<3, 0><<<grid, blk, 0, stream>>>(CONV_ARGS);
        else if (mode == 1)
            mvs_wmma_conv_kernel<3, 1><<<grid, blk, 0, stream>>>(CONV_ARGS);
        else
            mvs_wmma_conv_kernel<3, 2><<<grid, blk, 0, stream>>>(CONV_ARGS);
#undef CONV_ARGS
    };

    // homographies (independent of feature net)
    mvs_homog_kernel<<<dim3(3), dim3(64), 0, stream>>>(Kin, Ein, Hs);

    // feature net: conv+BN+ReLU x7, then conv+bias
    conv(imgs, bufA, cw[0], nullptr, cg[0], cb[0], cm[0], cvv[0],
         3,  3, 384, 512,  8, 384, 512, 3, 1, 1,  27,  27, 0);
    conv(bufA, bufB, cw[1], nullptr, cg[1], cb[1], cm[1], cvv[1],
         3,  8, 384, 512,  8, 384, 512, 3, 1, 1,  72,  72, 0);
    conv(bufB, bufA, cw[2], nullptr, cg[2], cb[2], cm[2], cvv[2],
         3,  8, 384, 512, 16, 192, 256, 5, 2, 2, 200, 200, 0);
    conv(bufA, bufB, cw[3], nullptr, cg[3], cb[3], cm[3], cvv[3],
         3, 16, 192, 256, 16, 192, 256, 3, 1, 1, 144, 144, 0);
    conv(bufB, bufA, cw[4], nullptr, cg[4], cb[4], cm[4], cvv[4],
         3, 16, 192, 256, 16, 192, 256, 3, 1, 1, 144, 144, 0);
    conv(bufA, bufB, cw[5], nullptr, cg[5], cb[5], cm[5], cvv[5],
         3, 16, 192, 256, 32,  96, 128, 5, 2, 2, 400, 400, 0);
    conv(bufB, bufA, cw[6], nullptr, cg[6], cb[6], cm[6], cvv[6],
         3, 32,  96, 128, 32,  96, 128, 3, 1, 1, 288, 288, 0);
    conv(bufA, feats, fw, nullptr, fb, nullptr, nullptr, nullptr,
         3, 32,  96, 128, 32,  96, 128, 3, 1, 1, 288, 288, 1);

    // plane-sweep warp + variance cost (stores -cost = reg_net input)
    mvs_warpcost_kernel<<<dim3((96 * 128 + 255) / 256, 48), dim3(256), 0, stream>>>(
        feats, Hs, cost);

    // reg net: GRUs collapse (h0 = 0) to sigmoid(conv_z) * tanh(conv_q);
    // only the x-channel slice of each GRU weight row is live (wstride = full row).
    conv(cost, g1, gwz[0], gwq[0], gbz[0], gbq[0], nullptr, nullptr,
         48, 32, 96, 128, 16, 96, 128, 3, 1, 1, 288, 432, 2);
    conv(g1, g2, gwz[1], gwq[1], gbz[1], gbq[1], nullptr, nullptr,
         48, 16, 96, 128,  4, 96, 128, 3, 1, 1, 144, 180, 2);
    conv(g2, g3, gwz[2], gwq[2], gbz[2], gbq[2], nullptr, nullptr,
         48,  4, 96, 128,  2, 96, 128, 3, 1, 1,  36,  54, 2);
    conv(g3, regout, pw, nullptr, pb, nullptr, nullptr, nullptr,
         48,  2, 96, 128,  1, 96, 128, 3, 1, 1,  18,  18, 1);

    // softmax over depth -> output (1,48,96,128)
    mvs_softmax_kernel<<<dim3((96 * 128 + 255) / 256), dim3(256), 0, stream>>>(
        regout, (float*)d_out);
}